// AttentionGNN_5317169512872
// MI455X (gfx1250) — compile-verified
//
#include <hip/hip_runtime.h>
#include <math.h>

typedef float v2f __attribute__((ext_vector_type(2)));
typedef float v8f __attribute__((ext_vector_type(8)));

#define GRAPHS   256
#define NPG      64
#define DEG      8
#define NNODES   16384
#define NEDGES   131072
#define NODE_IN  64
#define EDGE_IN  32
#define H        256
#define NLAYERS  4
#define NHEADS   8
#define DK       32

#if __has_builtin(__builtin_amdgcn_wmma_f32_16x16x4_f32)
#define HAVE_WMMA4 1
#endif

// ---------------------------------------------------------------------------
// One 16x16 f32 output tile, D = A(16xK) * B(Kx16), K multiple of 4.
// A: row-major, lda stride (LDS). B: row-major KxN, ldb stride (global).
// Frag layout per CDNA5 ISA 7.12.2:
//   A vgpr0/1: lane L holds A[M=L%16][K = kk + 2*(L/16) + {0,1}]
//   B vgpr0/1: lane L holds B[K = kk + 2*(L/16) + {0,1}][N=L%16]
//   D vgpr v : lane L holds D[M = v + 8*(L/16)][N = L%16]
// ---------------------------------------------------------------------------
__device__ __forceinline__ v8f wmma_tile(const float* A, int lda,
                                         const float* B, int ldb,
                                         int K, int l16, int hi2) {
  v8f acc = {};
#ifdef HAVE_WMMA4
  for (int kk = 0; kk < K; kk += 4) {
    v2f a, b;
    a.x = A[l16 * lda + kk + hi2];
    a.y = A[l16 * lda + kk + hi2 + 1];
    b.x = B[(kk + hi2) * ldb + l16];
    b.y = B[(kk + hi2 + 1) * ldb + l16];
    acc = __builtin_amdgcn_wmma_f32_16x16x4_f32(false, a, false, b,
                                                (short)0, acc, false, false);
  }
#else
  int hi = hi2 >> 1;
  for (int v = 0; v < 8; ++v) {
    float s = 0.f;
    for (int k = 0; k < K; ++k) s += A[(v + 8 * hi) * lda + k] * B[k * ldb + l16];
    acc[v] = s;
  }
#endif
  return acc;
}

// C[64 x Ncols] = A[64 x K] @ B[K x Ncols] (+bias)(+act). 8 waves cover tiles.
// act: 0 none, 1 relu, 2 exact gelu.
__device__ __forceinline__ void gemm64(const float* A, int lda,
                                       const float* B, int ldb,
                                       float* C, int ldc,
                                       int Ncols, int K, int tid,
                                       const float* bias, int act) {
  const int wave = tid >> 5, lane = tid & 31;
  const int l16 = lane & 15, hi = lane >> 4, hi2 = hi * 2;
  const int ntcols = Ncols >> 4;
  const int ntiles = 4 * ntcols;           // 64 rows -> 4 M-tiles
  for (int t = wave; t < ntiles; t += 8) {
    int mt = t / ntcols, nt = t % ntcols;
    v8f acc = wmma_tile(A + mt * 16 * lda, lda, B + nt * 16, ldb, K, l16, hi2);
    int col = nt * 16 + l16;
    float bv = bias ? bias[col] : 0.f;
#pragma unroll
    for (int v = 0; v < 8; ++v) {
      float x = acc[v] + bv;
      if (act == 1) x = fmaxf(x, 0.f);
      else if (act == 2) x = 0.5f * x * (1.f + erff(x * 0.70710678f));
      C[(mt * 16 + v + 8 * hi) * ldc + col] = x;
    }
  }
}

// In-place LayerNorm over H=256 of (xh + add), both in LDS. 4 threads/node.
__device__ __forceinline__ void ln_residual(float* xh, const float* add,
                                            const float* gamma, const float* beta,
                                            float eps, float* E_ln, float* E_mu,
                                            int tid) {
  const int n = tid >> 2, part = tid & 3, c0 = part * 64;
  float s = 0.f, s2 = 0.f;
  for (int j = 0; j < 64; ++j) {
    float v = xh[n * H + c0 + j] + add[n * H + c0 + j];
    s += v; s2 += v * v;
  }
  E_ln[n * 4 + part] = s;
  E_ln[256 + n * 4 + part] = s2;
  __syncthreads();
  if (tid < 64) {
    float ss = 0.f, qq = 0.f;
    for (int j = 0; j < 4; ++j) { ss += E_ln[tid * 4 + j]; qq += E_ln[256 + tid * 4 + j]; }
    float mu = ss * (1.f / 256.f);
    float var = qq * (1.f / 256.f) - mu * mu;
    E_mu[tid] = mu;
    E_mu[64 + tid] = rsqrtf(var + eps);
  }
  __syncthreads();
  float mu = E_mu[n], rs = E_mu[64 + n];
  for (int j = 0; j < 64; ++j) {
    int c = c0 + j;
    float v = xh[n * H + c] + add[n * H + c];
    xh[n * H + c] = (v - mu) * rs * gamma[c] + beta[c];
  }
  __syncthreads();
}

// ---------------------------------------------------------------------------
// Kernel 1: elog[i][e] = (edge_feats[e] @ We + be) . gat_a[i, 2H:3H]
//         = edge_feats[e] . (We @ a_e_i) + be . a_e_i
// Avoids materializing the 134MB edge embedding entirely.
// ---------------------------------------------------------------------------
__global__ __launch_bounds__(256) void elog_kernel(
    const float* __restrict__ ef, const float* __restrict__ We,
    const float* __restrict__ be, const float* __restrict__ gat_a,
    float* __restrict__ elog) {
  __shared__ float cvec[NLAYERS][EDGE_IN];
  __shared__ float dsc[NLAYERS];
  const int tid = threadIdx.x;
  if (tid < NLAYERS * EDGE_IN) {               // 128 threads: c vectors
    int i = tid >> 5, k = tid & 31;
    const float* ae = gat_a + i * 768 + 512;
    float s = 0.f;
    for (int j = 0; j < H; ++j) s += We[k * H + j] * ae[j];
    cvec[i][k] = s;
  }
  if (tid >= 128 && tid < 128 + NLAYERS) {     // 4 threads: scalar offsets
    int i = tid - 128;
    const float* ae = gat_a + i * 768 + 512;
    float s = 0.f;
    for (int j = 0; j < H; ++j) s += be[j] * ae[j];
    dsc[i] = s;
  }
  __syncthreads();
  const int e = blockIdx.x * 256 + tid;
  float f[EDGE_IN];
#pragma unroll
  for (int k = 0; k < EDGE_IN; ++k) f[k] = ef[e * EDGE_IN + k];
#pragma unroll
  for (int i = 0; i < NLAYERS; ++i) {
    float s = dsc[i];
#pragma unroll
    for (int k = 0; k < EDGE_IN; ++k) s += f[k] * cvec[i][k];
    elog[i * NEDGES + e] = s;
  }
}

// ---------------------------------------------------------------------------
// Kernel 2: fully-fused per-graph pipeline. block = graph, 256 threads.
// ---------------------------------------------------------------------------
__global__ __launch_bounds__(256) void graph_kernel(
    const float* __restrict__ node_feats, const int* __restrict__ dst,
    const float* __restrict__ Wn, const float* __restrict__ bn,
    const float* __restrict__ gat_W, const float* __restrict__ gat_a,
    const float* __restrict__ gat_lng, const float* __restrict__ gat_lnb,
    const float* __restrict__ Wq, const float* __restrict__ Wk,
    const float* __restrict__ Wv,
    const float* __restrict__ att_lng, const float* __restrict__ att_lnb,
    const float* __restrict__ ff_W1, const float* __restrict__ ff_b1,
    const float* __restrict__ ff_W2, const float* __restrict__ ff_b2,
    const float* __restrict__ ff_lng, const float* __restrict__ ff_lnb,
    const float* __restrict__ g_W1, const float* __restrict__ g_b1,
    const float* __restrict__ g_W2, const float* __restrict__ g_b2,
    const float* __restrict__ elog, float* __restrict__ out) {
  // 3 x 64KB activation buffers + 8KB scratch = ~202KB LDS (<320KB/WGP)
  __shared__ float S[3 * NPG * H + 2048];
  __shared__ int EDI[NPG * DEG];
  float* xh  = S;                  // [64][256] resident activations
  float* b1  = S + NPG * H;        // [64][256] (m / q / k / o / y-lo / z / r)
  float* b2  = S + 2 * NPG * H;    // [64][256] (v / y-hi)
  float* ext = S + 3 * NPG * H;
  float* E_red = ext;              // 512
  float* E_mu  = ext + 512;        // 128 (also s1/s2, gate scores)
  float* E_att = ext + 640;        // 512
  float* E_ln  = ext + 1152;       // 512

  const int g = blockIdx.x;
  const int tid = threadIdx.x;
  const int lane = tid & 31, l16 = lane & 15, hi = lane >> 4;

  // ---- stage 0: node_feats tile -> b1, project to h -> xh -----------------
  for (int idx = tid; idx < NPG * NODE_IN; idx += 256)
    b1[idx] = node_feats[(g * NPG + (idx >> 6)) * NODE_IN + (idx & 63)];
  __syncthreads();
  gemm64(b1, NODE_IN, Wn, H, xh, H, H, NODE_IN, tid, bn, 0);
  __syncthreads();

  // ---- stage 1: 4 GAT layers ---------------------------------------------
  for (int i = 0; i < NLAYERS; ++i) {
    const float* W  = gat_W + i * H * H;
    const float* a1 = gat_a + i * 768;
    const float* a2 = a1 + H;
    // m = xh @ W -> b1
    gemm64(xh, H, W, H, b1, H, H, H, tid, nullptr, 0);
    __syncthreads();
    // s1[n]=m[n].a1  s2[n]=m[n].a2  (4 threads per node)
    {
      int n = tid >> 2, c0 = (tid & 3) * 64;
      float p1 = 0.f, p2 = 0.f;
      for (int j = 0; j < 64; ++j) {
        float mv = b1[n * H + c0 + j];
        p1 += mv * a1[c0 + j];
        p2 += mv * a2[c0 + j];
      }
      E_red[tid] = p1; E_red[256 + tid] = p2;
    }
    __syncthreads();
    if (tid < 128) {
      int n = tid & 63, which = tid >> 6;
      float s = 0.f;
      for (int j = 0; j < 4; ++j) s += E_red[which * 256 + n * 4 + j];
      E_mu[tid] = s;                        // E_mu[n]=s1, E_mu[64+n]=s2
    }
    __syncthreads();
    // per-node segment softmax over its 8 consecutive edges
    if (tid < NPG) {
      int n = tid, eb = (g * NPG + n) * DEG;
      float lg[DEG]; float mx = -3.4e38f;
#pragma unroll
      for (int j = 0; j < DEG; ++j) {
        int dl = dst[eb + j] - g * NPG;
        EDI[n * DEG + j] = dl;
        float v = E_mu[n] + E_mu[64 + dl] + elog[i * NEDGES + eb + j];
        v = (v >= 0.f) ? v : 0.01f * v;     // LeakyReLU
        lg[j] = v; mx = fmaxf(mx, v);
      }
      float sm = 0.f;
#pragma unroll
      for (int j = 0; j < DEG; ++j) { float e = __expf(lg[j] - mx); E_att[n * DEG + j] = e; sm += e; }
      float inv = 1.f / sm;
#pragma unroll
      for (int j = 0; j < DEG; ++j) E_att[n * DEG + j] *= inv;
    }
    __syncthreads();
    // agg + residual + LayerNorm(eps=1e-5)
    {
      int n = tid >> 2, part = tid & 3, c0 = part * 64;
      float val[64];
#pragma unroll
      for (int j = 0; j < 64; ++j) val[j] = xh[n * H + c0 + j];
      for (int j = 0; j < DEG; ++j) {
        float w = E_att[n * DEG + j];
        const float* mr = b1 + EDI[n * DEG + j] * H + c0;
#pragma unroll
        for (int c = 0; c < 64; ++c) val[c] += w * mr[c];
      }
      float s = 0.f, s2 = 0.f;
#pragma unroll
      for (int j = 0; j < 64; ++j) { s += val[j]; s2 += val[j] * val[j]; }
      E_ln[n * 4 + part] = s; E_ln[256 + n * 4 + part] = s2;
      __syncthreads();
      if (tid < 64) {
        float ss = 0.f, qq = 0.f;
        for (int j = 0; j < 4; ++j) { ss += E_ln[tid * 4 + j]; qq += E_ln[256 + tid * 4 + j]; }
        float mu = ss * (1.f / 256.f);
        E_mu[tid] = mu;
        E_mu[64 + tid] = rsqrtf(qq * (1.f / 256.f) - mu * mu + 1e-5f);
      }
      __syncthreads();
      float mu = E_mu[n], rs = E_mu[64 + n];
      const float* gm = gat_lng + i * H; const float* bt = gat_lnb + i * H;
#pragma unroll
      for (int j = 0; j < 64; ++j) {
        int c = c0 + j;
        xh[n * H + c] = (val[j] - mu) * rs * gm[c] + bt[c];
      }
    }
    __syncthreads();
  }

  // ---- stage 2: global multi-head attention (flash-style) -----------------
  gemm64(xh, H, Wq, H, b1, H, H, H, tid, nullptr, 0);   // q -> b1
  __syncthreads();
  const int r1 = tid, r2 = tid + 256;                    // (head,query) rows
  const int hd1 = r1 >> 6, qi1 = r1 & 63, hd2 = r2 >> 6, qi2 = r2 & 63;
  float q1[DK], q2[DK];
#pragma unroll
  for (int d = 0; d < DK; ++d) { q1[d] = b1[qi1 * H + hd1 * DK + d];
                                 q2[d] = b1[qi2 * H + hd2 * DK + d]; }
  __syncthreads();
  gemm64(xh, H, Wk, H, b1, H, H, H, tid, nullptr, 0);   // k -> b1
  gemm64(xh, H, Wv, H, b2, H, H, H, tid, nullptr, 0);   // v -> b2
  __syncthreads();
  float o1[DK], o2[DK];
  const float scale = 0.17677669529663687f;              // 1/sqrt(32)
  {
    float m = -3.4e38f, den = 0.f;
#pragma unroll
    for (int d = 0; d < DK; ++d) o1[d] = 0.f;
    for (int j = 0; j < NPG; ++j) {
      const float* kr = b1 + j * H + hd1 * DK;
      float s = 0.f;
#pragma unroll
      for (int d = 0; d < DK; ++d) s += q1[d] * kr[d];
      s *= scale;
      float mn = fmaxf(m, s), corr = __expf(m - mn), p = __expf(s - mn);
      den = den * corr + p;
      const float* vr = b2 + j * H + hd1 * DK;
#pragma unroll
      for (int d = 0; d < DK; ++d) o1[d] = o1[d] * corr + p * vr[d];
      m = mn;
    }
    float inv = 1.f / den;
#pragma unroll
    for (int d = 0; d < DK; ++d) o1[d] *= inv;
  }
  {
    float m = -3.4e38f, den = 0.f;
#pragma unroll
    for (int d = 0; d < DK; ++d) o2[d] = 0.f;
    for (int j = 0; j < NPG; ++j) {
      const float* kr = b1 + j * H + hd2 * DK;
      float s = 0.f;
#pragma unroll
      for (int d = 0; d < DK; ++d) s += q2[d] * kr[d];
      s *= scale;
      float mn = fmaxf(m, s), corr = __expf(m - mn), p = __expf(s - mn);
      den = den * corr + p;
      const float* vr = b2 + j * H + hd2 * DK;
#pragma unroll
      for (int d = 0; d < DK; ++d) o2[d] = o2[d] * corr + p * vr[d];
      m = mn;
    }
    float inv = 1.f / den;
#pragma unroll
    for (int d = 0; d < DK; ++d) o2[d] *= inv;
  }
  __syncthreads();                                       // all reads of k/v done
#pragma unroll
  for (int d = 0; d < DK; ++d) { b1[qi1 * H + hd1 * DK + d] = o1[d];
                                 b1[qi2 * H + hd2 * DK + d] = o2[d]; }
  __syncthreads();
  ln_residual(xh, b1, att_lng, att_lnb, 1e-6f, E_ln, E_mu, tid);

  // ---- stage 3: feed-forward ----------------------------------------------
  // y = gelu(xh @ W1 + b1)  [64 x 512] spans b1..b2
  gemm64(xh, H, ff_W1, 2 * H, b1, 2 * H, 2 * H, H, tid, ff_b1, 2);
  __syncthreads();
  {
    // z = y @ W2 + b2 -> registers first (z overwrites y), then store to b1
    const int wave = tid >> 5, hi2 = hi * 2;
    v8f accs[8];
    for (int ti = 0; ti < 8; ++ti) {
      int t = wave + ti * 8, mt = t >> 4, nt = t & 15;
      accs[ti] = wmma_tile(b1 + mt * 16 * (2 * H), 2 * H,
                           ff_W2 + nt * 16, H, 2 * H, l16, hi2);
    }
    __syncthreads();
    for (int ti = 0; ti < 8; ++ti) {
      int t = wave + ti * 8, mt = t >> 4, nt = t & 15;
      int col = nt * 16 + l16;
#pragma unroll
      for (int v = 0; v < 8; ++v)
        b1[(mt * 16 + v + 8 * hi) * H + col] = accs[ti][v] + ff_b2[col];
    }
  }
  __syncthreads();
  ln_residual(xh, b1, ff_lng, ff_lnb, 1e-6f, E_ln, E_mu, tid);

  // ---- stage 4: gating readout --------------------------------------------
  gemm64(xh, H, g_W1, H, b1, H, H, H, tid, g_b1, 1);     // relu(x@W1+b1)
  __syncthreads();
  {
    int n = tid >> 2, c0 = (tid & 3) * 64;
    float p = 0.f;
    for (int j = 0; j < 64; ++j) p += b1[n * H + c0 + j] * g_W2[c0 + j];
    E_red[tid] = p;
  }
  __syncthreads();
  if (tid < 64) {
    float s = g_b2[0];
    for (int j = 0; j < 4; ++j) s += E_red[tid * 4 + j];
    E_mu[tid] = s;
  }
  __syncthreads();
  if (tid < 64) {
    float mx = -3.4e38f;
    for (int n = 0; n < NPG; ++n) mx = fmaxf(mx, E_mu[n]);
    E_att[tid] = __expf(E_mu[tid] - mx);
  }
  __syncthreads();
  {
    int c = tid;
    float den = 0.f, acc = 0.f;
    for (int n = 0; n < NPG; ++n) { den += E_att[n]; acc += E_att[n] * xh[n * H + c]; }
    out[g * H + c] = acc / den;
  }
}

// ---------------------------------------------------------------------------
extern "C" void kernel_launch(void* const* d_in, const int* in_sizes, int n_in,
                              void* d_out, int out_size, void* d_ws, size_t ws_size,
                              hipStream_t stream) {
  const float* node_feats = (const float*)d_in[0];
  const float* edge_feats = (const float*)d_in[1];
  const int*   dst        = (const int*)d_in[3];
  const float* Wn      = (const float*)d_in[4];
  const float* bn      = (const float*)d_in[5];
  const float* We      = (const float*)d_in[6];
  const float* be      = (const float*)d_in[7];
  const float* gat_W   = (const float*)d_in[8];
  const float* gat_a   = (const float*)d_in[9];
  const float* gat_lng = (const float*)d_in[10];
  const float* gat_lnb = (const float*)d_in[11];
  const float* Wq      = (const float*)d_in[12];
  const float* Wk      = (const float*)d_in[13];
  const float* Wv      = (const float*)d_in[14];
  const float* att_lng = (const float*)d_in[15];
  const float* att_lnb = (const float*)d_in[16];
  const float* ff_W1   = (const float*)d_in[17];
  const float* ff_b1   = (const float*)d_in[18];
  const float* ff_W2   = (const float*)d_in[19];
  const float* ff_b2   = (const float*)d_in[20];
  const float* ff_lng  = (const float*)d_in[21];
  const float* ff_lnb  = (const float*)d_in[22];
  const float* g_W1    = (const float*)d_in[23];
  const float* g_b1    = (const float*)d_in[24];
  const float* g_W2    = (const float*)d_in[25];
  const float* g_b2    = (const float*)d_in[26];

  float* elog = (float*)d_ws;  // [NLAYERS][NEDGES] = 2MB

  elog_kernel<<<NEDGES / 256, 256, 0, stream>>>(edge_feats, We, be, gat_a, elog);
  graph_kernel<<<GRAPHS, 256, 0, stream>>>(
      node_feats, dst, Wn, bn, gat_W, gat_a, gat_lng, gat_lnb,
      Wq, Wk, Wv, att_lng, att_lnb, ff_W1, ff_b1, ff_W2, ff_b2,
      ff_lng, ff_lnb, g_W1, g_b1, g_W2, g_b2, elog, (float*)d_out);
}